// Net_14620068675811
// MI455X (gfx1250) — compile-verified
//
#include <hip/hip_runtime.h>
#include <stdint.h>

// SNN leaky scan: spk[T,B,N] from x[T,B,N]; recurrence sequential in T,
// elementwise over B*N. Pure HBM-streaming problem (no matmul -> no WMMA).
// CDNA5 path: global_load_async_to_lds_b128 triple-buffered pipeline
// (ASYNCcnt, in-order retirement), ds_load_b128 reads, non-temporal spike
// stores so the 128 MiB input stays resident in the 192 MB L2 across replays.

typedef float v4f __attribute__((ext_vector_type(4)));

#define T_STEPS 128
#define PF      8                  // time-slabs per chunk
#define NCHUNK  (T_STEPS / PF)     // 16 chunks
#define NBUF    3                  // triple buffer: 2 chunks in flight while computing
#define BLK     256                // threads per block (8 wave32)

// Async copy 16B/lane from global to LDS (ASYNCcnt path, no VGPR destination).
__device__ __forceinline__ void async_ld_b128(uint32_t lds_off, uint64_t gaddr) {
    asm volatile("global_load_async_to_lds_b128 %0, %1, off"
                 :: "v"(lds_off), "v"(gaddr)
                 : "memory");
}

// Async loads retire in order: ASYNCcnt <= N  =>  all but the newest N landed.
__device__ __forceinline__ void wait_async_16() {
    asm volatile("s_wait_asynccnt 16" ::: "memory");
}
__device__ __forceinline__ void wait_async_8() {
    asm volatile("s_wait_asynccnt 8" ::: "memory");
}
__device__ __forceinline__ void wait_async_0() {
    asm volatile("s_wait_asynccnt 0" ::: "memory");
}

// One LIF step on 4 neurons: mem = fma(0.5, mem, x) - reset; spk = H(mem - 1).
// reset_t == spk_{t-1}, so `rst` carries the previous spike. 0.5*mem is exact,
// so fmaf matches the reference's mul+add rounding.
__device__ __forceinline__ void lif_step(v4f& mem, v4f& rst, const v4f xv) {
    mem.x = __builtin_fmaf(0.5f, mem.x, xv.x) - rst.x;
    mem.y = __builtin_fmaf(0.5f, mem.y, xv.y) - rst.y;
    mem.z = __builtin_fmaf(0.5f, mem.z, xv.z) - rst.z;
    mem.w = __builtin_fmaf(0.5f, mem.w, xv.w) - rst.w;
    rst.x = mem.x > 1.0f ? 1.0f : 0.0f;
    rst.y = mem.y > 1.0f ? 1.0f : 0.0f;
    rst.z = mem.z > 1.0f ? 1.0f : 0.0f;
    rst.w = mem.w > 1.0f ? 1.0f : 0.0f;
}

// Issue one chunk (PF time slabs) of async loads into an LDS buffer.
__device__ __forceinline__ void issue_chunk(uint32_t lds_lane0, uint32_t buf_off,
                                            uint64_t gsrc, uint64_t step_b, int tbase) {
#pragma unroll
    for (int k = 0; k < PF; ++k)
        async_ld_b128(lds_lane0 + buf_off + (uint32_t)k * (BLK * (uint32_t)sizeof(v4f)),
                      gsrc + (uint64_t)(tbase + k) * step_b);
}

// Consume one staged chunk: PF LIF steps + NT spike stores.
__device__ __forceinline__ void compute_chunk(const v4f (&buf)[PF][BLK], int tid,
                                              v4f& mem, v4f& rst,
                                              v4f* __restrict__ dst, int tbase, int nvec) {
#pragma unroll
    for (int k = 0; k < PF; ++k) {
        const v4f xv = buf[k][tid];               // ds_load_b128, own lane's data
        lif_step(mem, rst, xv);
        // NT store: spikes are write-once; keep them out of L2 so x stays resident.
        __builtin_nontemporal_store(rst, dst + (size_t)(tbase + k) * (size_t)nvec);
    }
}

__global__ __launch_bounds__(BLK) void snn_leaky_scan(
        const float* __restrict__ xin, float* __restrict__ spk, int nvec) {
    // Triple-buffered LDS staging: 3 chunks x PF slabs x BLK lanes x 16B = 96 KiB.
    __shared__ v4f lds[NBUF][PF][BLK];

    const int tid = threadIdx.x;
    const int idx = blockIdx.x * BLK + tid;       // float4 index within a time slab
    if (idx >= nvec) return;

    const uint64_t step_b = (uint64_t)nvec * sizeof(v4f);   // bytes per time step (1 MiB)
    const uint64_t gsrc   = (uint64_t)(uintptr_t)xin + (uint64_t)idx * sizeof(v4f);
    v4f* __restrict__ dst = (v4f*)spk + idx;

    // Low 32 bits of the generic shared address == LDS byte offset (ISA aperture).
    const uint32_t lds_lane0 = (uint32_t)(uintptr_t)(&lds[0][0][tid]);
    const uint32_t buf_b     = PF * BLK * (uint32_t)sizeof(v4f);   // 32 KiB per chunk

    // Prologue: chunks 0 and 1 in flight (16 async loads).
    issue_chunk(lds_lane0, 0u,    gsrc, step_b, 0);
    issue_chunk(lds_lane0, buf_b, gsrc, step_b, PF);

    v4f mem = {0.f, 0.f, 0.f, 0.f};
    v4f rst = {0.f, 0.f, 0.f, 0.f};   // spk_{t-1} == reset_t ; H(mem0 - thr) = 0

    int cur = 0;        // buffer holding the chunk to compute
    int nxt = 2;        // buffer to fill with chunk c+2
    for (int c = 0; c < NCHUNK - 2; ++c) {        // c = 0..13
        issue_chunk(lds_lane0, (uint32_t)nxt * buf_b, gsrc, step_b, (c + 2) * PF);
        wait_async_16();                          // chunks c+1, c+2 may pend => chunk c landed
        compute_chunk(lds[cur], tid, mem, rst, dst, c * PF, nvec);
        cur = (cur + 1 == NBUF) ? 0 : cur + 1;
        nxt = (nxt + 1 == NBUF) ? 0 : nxt + 1;
    }

    // Drain: chunk 14 then chunk 15 (no further issues).
    wait_async_8();
    compute_chunk(lds[cur], tid, mem, rst, dst, (NCHUNK - 2) * PF, nvec);
    cur = (cur + 1 == NBUF) ? 0 : cur + 1;

    wait_async_0();
    compute_chunk(lds[cur], tid, mem, rst, dst, (NCHUNK - 1) * PF, nvec);
}

extern "C" void kernel_launch(void* const* d_in, const int* in_sizes, int n_in,
                              void* d_out, int out_size, void* d_ws, size_t ws_size,
                              hipStream_t stream) {
    (void)n_in; (void)out_size; (void)d_ws; (void)ws_size;
    const float* x = (const float*)d_in[0];
    float* out     = (float*)d_out;

    const int total = in_sizes[0];        // T*B*N = 128*64*4096
    const int bn    = total / T_STEPS;    // 262144 neurons
    const int nvec  = bn / 4;             // 65536 float4 lanes per time step

    dim3 block(BLK);
    dim3 grid((nvec + BLK - 1) / BLK);    // 256 blocks = 2048 wave32
    snn_leaky_scan<<<grid, block, 0, stream>>>(x, out, nvec);
}